// NeuralNetwork_5274219839793
// MI455X (gfx1250) — compile-verified
//
#include <hip/hip_runtime.h>
#include <math.h>

// Problem sizes (fixed by reference)
#define BB   64
#define TTT  128
#define DDD  64
#define HHH  256
#define G3   768                 // 3*H
#define DL   (TTT * HHH)         // 32768 per batch row
#define NBD  ((size_t)BB * DL)   // 2,097,152 floats per ring slot
#define MAND 6
#define LAMR 1e-4f
#define EPSR 1e-3f

typedef float v2f __attribute__((ext_vector_type(2)));
typedef float v8f __attribute__((ext_vector_type(8)));
typedef unsigned int v4u __attribute__((ext_vector_type(4)));
typedef int v4i __attribute__((ext_vector_type(4)));
typedef int v8i __attribute__((ext_vector_type(8)));

#if defined(__HIP_DEVICE_COMPILE__) && __has_builtin(__builtin_amdgcn_tensor_load_to_lds) && __has_builtin(__builtin_amdgcn_s_wait_tensorcnt)
#define HAVE_TDM 1
#else
#define HAVE_TDM 0
#endif

// ---------------------------------------------------------------------------
// f32 WMMA 16x16x4 (exact fp32 semantics on the matrix pipe).
// A[m][k]: lane l holds A[l&15][k0+2*(l>>4)+{0,1}]
// B[k][n]: lane l holds B[k0+2*(l>>4)+{0,1}][l&15]
// D[m][n]: vgpr=m&7, lane=(m>>3)*16+n
// ---------------------------------------------------------------------------
__device__ __forceinline__ v8f wmma_f32_k4(v2f a, v2f b, v8f c) {
#if __HIP_DEVICE_COMPILE__
  return __builtin_amdgcn_wmma_f32_16x16x4_f32(false, a, false, b, (short)0, c,
                                               false, false);
#else
  return c;
#endif
}

#if HAVE_TDM
// ---------------------------------------------------------------------------
// TDM: async DMA of a 2D f32 tile [rows x cols] (global, row stride
// row_stride_elems) into LDS at byte offset lds_off.  D# per CDNA5 ISA §8.
// ---------------------------------------------------------------------------
__device__ __forceinline__ void tdm_load_2d(unsigned lds_off, const void* gptr,
                                            unsigned rows, unsigned cols,
                                            unsigned row_stride_elems) {
  unsigned long long ga = (unsigned long long)gptr;
  v4u g0;
  g0.x = 1u;                                   // count=1 (valid), user mode
  g0.y = lds_off;                              // lds_addr (bytes)
  g0.z = (unsigned)ga;                         // global_addr[31:0]
  g0.w = (unsigned)(ga >> 32) | (2u << 30);    // global_addr[56:32] | type=2
  v8i g1;
  g1[0] = (int)(2u << 16);                     // data_size=2 (4 bytes); mask=0
  g1[1] = (int)(cols << 16);                   // tensor_dim0[15:0] @ bits 63:48
  g1[2] = (int)((cols >> 16) | (rows << 16));  // dim0[31:16] | tensor_dim1[15:0]
  g1[3] = (int)((rows >> 16) | (cols << 16));  // dim1[31:16] | tile_dim0
  g1[4] = (int)rows;                           // tile_dim1 (tile_dim2=0)
  g1[5] = (int)row_stride_elems;               // tensor_dim0_stride[31:0]
  g1[6] = 0;                                   // stride[47:32] | dim1_stride lo
  g1[7] = 0;
  v4i gz = {0, 0, 0, 0};
#if defined(__clang_major__) && (__clang_major__ >= 23)
  v8i gz8 = {0, 0, 0, 0, 0, 0, 0, 0};
  __builtin_amdgcn_tensor_load_to_lds(g0, g1, gz, gz, gz8, 0);
#else
  __builtin_amdgcn_tensor_load_to_lds(g0, g1, gz, gz, 0);
#endif
}
__device__ __forceinline__ void tdm_wait() {
  __builtin_amdgcn_s_wait_tensorcnt(0);
}
#endif

// ---------------------------------------------------------------------------
// transpose: out[c][r] = in[r][c]   (in: rows x cols)
// ---------------------------------------------------------------------------
__global__ void transpose_kernel(const float* __restrict__ in,
                                 float* __restrict__ out, int rows, int cols) {
  int i = blockIdx.x * blockDim.x + threadIdx.x;
  if (i < rows * cols) {
    int r = i / cols, c = i % cols;
    out[(size_t)c * rows + r] = in[(size_t)r * cols + c];
  }
}

// ---------------------------------------------------------------------------
// GI[r, 0..767] = Z[r, :K] @ W^T + bias, via WT[K][768].  One wave per 16x16
// tile, 8 waves per block. R multiple of 16.
// ---------------------------------------------------------------------------
__global__ void gemm_in_kernel(const float* __restrict__ Z,
                               const float* __restrict__ WT,
                               const float* __restrict__ bias,
                               float* __restrict__ GI, int R, int K) {
  int wid  = blockIdx.x * (blockDim.x >> 5) + (threadIdx.x >> 5);
  int tr   = wid / 48, tc = wid % 48;
  if (tr * 16 >= R) return;                  // wave-uniform
  int lane = threadIdx.x & 31, half = lane >> 4, lq = lane & 15;
  int r0 = tr * 16, c0 = tc * 16;

  v8f acc = {0.f, 0.f, 0.f, 0.f, 0.f, 0.f, 0.f, 0.f};
  for (int k0 = 0; k0 < K; k0 += 4) {
    int ka = k0 + 2 * half;
    v2f a, b;
    const float* ap = Z + (size_t)(r0 + lq) * K + ka;
    a.x = ap[0];  a.y = ap[1];
    const float* bp = WT + (size_t)ka * G3 + c0 + lq;
    b.x = bp[0];  b.y = bp[G3];
    acc = wmma_f32_k4(a, b, acc);
  }
  float bb = bias[c0 + lq];
#pragma unroll
  for (int i = 0; i < 8; ++i) {
    int m = i + 8 * half;
    GI[(size_t)(r0 + m) * G3 + c0 + lq] = acc[i] + bb;
  }
}

// ---------------------------------------------------------------------------
// GRU scan over T timesteps. 4 blocks x 512 threads; block owns 16 batch rows,
// h[16][256] in LDS. Per step:
//   - wave0 issues a TDM async tile load of this step's GI [16x768] into LDS
//     (overlaps with the WMMA recurrent GEMM below)
//   - gh = h @ Whh^T (+bhh): 16 waves x 3 adjacent N-tiles; K chunked so each
//     A-fragment batch is loaded from LDS once and reused by 3 accumulators
//   - s_wait_tensorcnt + barrier, then fused gates reading GI from LDS
// addTanh=0: OUT = h (x-GRU producing z0).  addTanh=1: OUT = tanh(h + z0).
// ---------------------------------------------------------------------------
__global__ void gru_scan_kernel(const float* __restrict__ GI,    // [B*T,768]
                                const float* __restrict__ WhhT,  // [256,768]
                                const float* __restrict__ bhh,   // [768]
                                const float* __restrict__ z0in,  // [B*T,256]|null
                                float* __restrict__ OUT,         // [B*T,256]
                                int addTanh) {
  __shared__ float hS[16][HHH];
  __shared__ float ghS[16][G3];
  __shared__ float giS[16][G3];
  const int b0 = blockIdx.x * 16;
  const int tid = threadIdx.x;
  const int wave = tid >> 5;
  const int lane = tid & 31, half = lane >> 4, lq = lane & 15;
  const int c0 = wave * 48;                  // 3 tiles: c0, c0+16, c0+32

  for (int i = tid; i < 16 * HHH; i += blockDim.x) (&hS[0][0])[i] = 0.f;
  __syncthreads();

  for (int t = 0; t < TTT; ++t) {
#if HAVE_TDM
    if (wave == 0) {
      tdm_load_2d((unsigned)(uintptr_t)&giS[0][0],
                  GI + ((size_t)b0 * TTT + t) * G3,
                  16u, (unsigned)G3, (unsigned)(TTT * G3));
    }
#endif
    // ---- recurrent GEMM on matrix cores, A-fragments cached per K-chunk ----
    v8f acc0 = {0.f, 0.f, 0.f, 0.f, 0.f, 0.f, 0.f, 0.f};
    v8f acc1 = acc0, acc2 = acc0;
    for (int kc = 0; kc < HHH; kc += 32) {
      v2f af[8];
#pragma unroll
      for (int j = 0; j < 8; ++j) {
        int ka = kc + 4 * j + 2 * half;
        af[j].x = hS[lq][ka];
        af[j].y = hS[lq][ka + 1];
      }
#pragma unroll
      for (int j = 0; j < 8; ++j) {
        int ka = kc + 4 * j + 2 * half;
        const float* bp = WhhT + (size_t)ka * G3 + lq;
        v2f bv0, bv1, bv2;
        bv0.x = bp[c0];       bv0.y = bp[c0 + G3];
        bv1.x = bp[c0 + 16];  bv1.y = bp[c0 + 16 + G3];
        bv2.x = bp[c0 + 32];  bv2.y = bp[c0 + 32 + G3];
        acc0 = wmma_f32_k4(af[j], bv0, acc0);
        acc1 = wmma_f32_k4(af[j], bv1, acc1);
        acc2 = wmma_f32_k4(af[j], bv2, acc2);
      }
    }
    {
      float bb0 = bhh[c0 + lq];
      float bb1 = bhh[c0 + 16 + lq];
      float bb2 = bhh[c0 + 32 + lq];
#pragma unroll
      for (int i = 0; i < 8; ++i) {
        int m = i + 8 * half;
        ghS[m][c0 + lq]      = acc0[i] + bb0;
        ghS[m][c0 + 16 + lq] = acc1[i] + bb1;
        ghS[m][c0 + 32 + lq] = acc2[i] + bb2;
      }
    }
#if HAVE_TDM
    if (wave == 0) tdm_wait();
#endif
    __syncthreads();
    // ---- fused gates ----
    for (int e = tid; e < 16 * HHH; e += blockDim.x) {
      int m = e >> 8, hc = e & 255;
      size_t row = (size_t)(b0 + m) * TTT + t;
#if HAVE_TDM
      float ir  = giS[m][hc];
      float iz  = giS[m][hc + 256];
      float inn = giS[m][hc + 512];
#else
      const float* gi = GI + row * G3;
      float ir = gi[hc], iz = gi[hc + 256], inn = gi[hc + 512];
#endif
      float hr = ghS[m][hc], hz = ghS[m][hc + 256], hn = ghS[m][hc + 512];
      float r  = 1.f / (1.f + __expf(-(ir + hr)));
      float zg = 1.f / (1.f + __expf(-(iz + hz)));
      float nn = tanhf(inn + r * hn);
      float hnew = (1.f - zg) * nn + zg * hS[m][hc];
      hS[m][hc] = hnew;
      float o = hnew;
      if (addTanh) o = tanhf(hnew + z0in[row * HHH + hc]);
      OUT[row * HHH + hc] = o;
    }
    __syncthreads();
  }
}

// ---------------------------------------------------------------------------
// Anderson alpha: per batch b, GGT over slots 0..n-1 (G=F-X), bordered
// (n+1)x(n+1) solve (Gauss-Jordan, partial pivot), alpha[b][0..n-1].
// Rings slot-major: slot i at ptr + i*NBD, row b at + b*DL.
// ---------------------------------------------------------------------------
__global__ void alpha_kernel(const float* __restrict__ X,
                             const float* __restrict__ F,
                             float* __restrict__ alpha, int n) {
  int b = blockIdx.x, tid = threadIdx.x;
  float acc[21];
#pragma unroll
  for (int p = 0; p < 21; ++p) acc[p] = 0.f;
  for (int d = tid; d < DL; d += 256) {
    float g[MAND];
    for (int i = 0; i < n; ++i) {
      size_t off = (size_t)i * NBD + (size_t)b * DL + d;
      g[i] = F[off] - X[off];
    }
    int p = 0;
    for (int i = 0; i < n; ++i)
      for (int j = 0; j <= i; ++j) acc[p++] += g[i] * g[j];
  }
  __shared__ float sm[256];
  __shared__ float GG[21];
  int np = n * (n + 1) / 2;
  for (int p = 0; p < np; ++p) {
    sm[tid] = acc[p];
    __syncthreads();
    for (int s = 128; s > 0; s >>= 1) {
      if (tid < s) sm[tid] += sm[tid + s];
      __syncthreads();
    }
    if (tid == 0) GG[p] = sm[0];
    __syncthreads();
  }
  if (tid == 0) {
    float GGf[MAND][MAND];
    int p = 0;
    for (int i = 0; i < n; ++i)
      for (int j = 0; j <= i; ++j) { GGf[i][j] = GG[p]; GGf[j][i] = GG[p]; ++p; }
    int m = n + 1;
    float Hm[7][8];
    for (int i = 0; i < m; ++i) {
      for (int j = 0; j < m; ++j) {
        float v;
        if (i == 0 && j == 0) v = 0.f;
        else if (i == 0 || j == 0) v = 1.f;
        else v = GGf[i - 1][j - 1] + ((i == j) ? LAMR : 0.f);
        Hm[i][j] = v;
      }
      Hm[i][m] = (i == 0) ? 1.f : 0.f;
    }
    for (int c = 0; c < m; ++c) {
      int piv = c; float pv = fabsf(Hm[c][c]);
      for (int r2 = c + 1; r2 < m; ++r2) {
        float v = fabsf(Hm[r2][c]);
        if (v > pv) { pv = v; piv = r2; }
      }
      if (piv != c)
        for (int j = 0; j <= m; ++j) { float tv = Hm[c][j]; Hm[c][j] = Hm[piv][j]; Hm[piv][j] = tv; }
      float inv = 1.f / Hm[c][c];
      for (int r2 = 0; r2 < m; ++r2) {
        if (r2 == c) continue;
        float f2 = Hm[r2][c] * inv;
        for (int j = c; j <= m; ++j) Hm[r2][j] -= f2 * Hm[c][j];
      }
    }
    for (int i = 0; i < n; ++i) alpha[b * MAND + i] = Hm[i + 1][m] / Hm[i + 1][i + 1];
  }
}

// Xnew(slot) = sum_i alpha[b][i] * F[i]     (BETA == 1.0 -> X term vanishes)
__global__ void xnew_kernel(float* __restrict__ X, const float* __restrict__ F,
                            const float* __restrict__ alpha, int n, int slot) {
  size_t idx = (size_t)blockIdx.x * blockDim.x + threadIdx.x;
  if (idx >= NBD) return;
  int b = (int)(idx / DL);
  float s = 0.f;
  for (int i = 0; i < n; ++i) s += alpha[b * MAND + i] * F[(size_t)i * NBD + idx];
  X[(size_t)slot * NBD + idx] = s;
}

// deterministic two-stage ||F-X||^2: 256 contiguous chunks of 8192
__global__ void diff1_kernel(const float* __restrict__ Fs,
                             const float* __restrict__ Xs,
                             float* __restrict__ partials) {
  __shared__ float sm[256];
  size_t base = (size_t)blockIdx.x * 8192;
  float s = 0.f;
  for (int i = threadIdx.x; i < 8192; i += 256) {
    float d = Fs[base + i] - Xs[base + i];
    s += d * d;
  }
  sm[threadIdx.x] = s;
  __syncthreads();
  for (int st = 128; st > 0; st >>= 1) {
    if (threadIdx.x < st) sm[threadIdx.x] += sm[threadIdx.x + st];
    __syncthreads();
  }
  if (threadIdx.x == 0) partials[blockIdx.x] = sm[0];
}

// single-block control: diff=sqrt(sum); improved/stopped/lowest bookkeeping
__global__ void ctrl_kernel(const float* __restrict__ partials,
                            float* __restrict__ scal,  // [0]=lowest
                            int* __restrict__ flags) { // [0]=stopped [1]=improved
  __shared__ float sm[256];
  int tid = threadIdx.x;
  sm[tid] = partials[tid];
  __syncthreads();
  for (int st = 128; st > 0; st >>= 1) {
    if (tid < st) sm[tid] += sm[tid + st];
    __syncthreads();
  }
  if (tid == 0) {
    float diff = sqrtf(sm[0]);
    int active = !flags[0];
    int improved = active && (diff < scal[0]);
    if (improved) scal[0] = diff;
    flags[1] = improved;
    if (active && diff < EPSR) flags[0] = 1;
  }
}

__global__ void copy_kernel(const float* __restrict__ src, float* __restrict__ dst,
                            int n, const int* __restrict__ flag) {
  if (flag != nullptr && *flag == 0) return;
  int i = blockIdx.x * blockDim.x + threadIdx.x;
  if (i < n) dst[i] = src[i];
}

__global__ void init_kernel(float* scal, int* flags) {
  if (threadIdx.x == 0) { scal[0] = 1e8f; flags[0] = 0; flags[1] = 0; }
}

// out[b] = dot(Z[b, :, :].flat(T-major), Wout) + bout
__global__ void out_kernel(const float* __restrict__ Z,
                           const float* __restrict__ Wout,
                           const float* __restrict__ bout,
                           float* __restrict__ out) {
  __shared__ float sm[256];
  int b = blockIdx.x, tid = threadIdx.x;
  float s = 0.f;
  for (int i = tid; i < DL; i += 256) s += Z[(size_t)b * DL + i] * Wout[i];
  sm[tid] = s;
  __syncthreads();
  for (int st = 128; st > 0; st >>= 1) {
    if (tid < st) sm[tid] += sm[tid + st];
    __syncthreads();
  }
  if (tid == 0) out[b] = sm[0] + bout[0];
}

// ---------------------------------------------------------------------------
static void f_eval(const float* Zin, float* Fout, const float* WihZT,
                   const float* bih_z, const float* WhhZT, const float* bhh_z,
                   const float* z0, float* GI, hipStream_t stream) {
  gemm_in_kernel<<<3072, 256, 0, stream>>>(Zin, WihZT, bih_z, GI, BB * TTT, HHH);
  gru_scan_kernel<<<4, 512, 0, stream>>>(GI, WhhZT, bhh_z, z0, Fout, 1);
}

extern "C" void kernel_launch(void* const* d_in, const int* in_sizes, int n_in,
                              void* d_out, int out_size, void* d_ws, size_t ws_size,
                              hipStream_t stream) {
  const float* x     = (const float*)d_in[0];   // [64,128,64]
  const float* Wih_x = (const float*)d_in[1];   // [768,64]
  const float* Whh_x = (const float*)d_in[2];   // [768,256]
  const float* bih_x = (const float*)d_in[3];
  const float* bhh_x = (const float*)d_in[4];
  const float* Wih_z = (const float*)d_in[5];   // [768,256]
  const float* Whh_z = (const float*)d_in[6];   // [768,256]
  const float* bih_z = (const float*)d_in[7];
  const float* bhh_z = (const float*)d_in[8];
  const float* Wout  = (const float*)d_in[9];   // [1,32768]
  const float* bout  = (const float*)d_in[10];  // [1]
  float* out = (float*)d_out;

  float* ws = (float*)d_ws;
  size_t o = 0;
  float* z0    = ws + o; o += NBD;                     // 2,097,152
  float* GI    = ws + o; o += (size_t)BB * TTT * G3;   // 6,291,456
  float* WhhXT = ws + o; o += (size_t)HHH * G3;
  float* WhhZT = ws + o; o += (size_t)HHH * G3;
  float* WihXT = ws + o; o += (size_t)DDD * G3;
  float* WihZT = ws + o; o += (size_t)HHH * G3;
  float* Xr    = ws + o; o += (size_t)MAND * NBD;
  float* Fr    = ws + o; o += (size_t)MAND * NBD;
  float* best  = ws + o; o += NBD;
  float* alpha = ws + o; o += 512;
  float* parts = ws + o; o += 256;
  float* scal  = ws + o; o += 4;
  int*   flags = (int*)(ws + o);

  const int N2 = (int)NBD;                    // 2,097,152
  const int CPB = 8192;                       // copy blocks

  // ---- prologue: transposed weights, scalar state ----
  transpose_kernel<<<(G3 * HHH + 255) / 256, 256, 0, stream>>>(Whh_x, WhhXT, G3, HHH);
  transpose_kernel<<<(G3 * HHH + 255) / 256, 256, 0, stream>>>(Whh_z, WhhZT, G3, HHH);
  transpose_kernel<<<(G3 * DDD + 255) / 256, 256, 0, stream>>>(Wih_x, WihXT, G3, DDD);
  transpose_kernel<<<(G3 * HHH + 255) / 256, 256, 0, stream>>>(Wih_z, WihZT, G3, HHH);
  init_kernel<<<1, 32, 0, stream>>>(scal, flags);

  // ---- z0 = gru_x(x) ----
  gemm_in_kernel<<<3072, 256, 0, stream>>>(x, WihXT, bih_x, GI, BB * TTT, DDD);
  gru_scan_kernel<<<4, 512, 0, stream>>>(GI, WhhXT, bhh_x, nullptr, z0, 0);

  // ---- Anderson init: X0=x0, best=x0, F0=f(x0), X1=F0, F1=f(F0) ----
  copy_kernel<<<CPB, 256, 0, stream>>>(z0, Xr, N2, nullptr);
  copy_kernel<<<CPB, 256, 0, stream>>>(z0, best, N2, nullptr);
  f_eval(Xr, Fr, WihZT, bih_z, WhhZT, bhh_z, z0, GI, stream);
  copy_kernel<<<CPB, 256, 0, stream>>>(Fr, Xr + NBD, N2, nullptr);
  f_eval(Xr + NBD, Fr + NBD, WihZT, bih_z, WhhZT, bhh_z, z0, GI, stream);

  // ---- Anderson iterations k = 2..49 ----
  for (int k = 2; k < 50; ++k) {
    int n = (k < MAND) ? k : MAND;
    int slot = k % MAND;
    alpha_kernel<<<BB, 256, 0, stream>>>(Xr, Fr, alpha, n);
    xnew_kernel<<<CPB, 256, 0, stream>>>(Xr, Fr, alpha, n, slot);
    f_eval(Xr + (size_t)slot * NBD, Fr + (size_t)slot * NBD,
           WihZT, bih_z, WhhZT, bhh_z, z0, GI, stream);
    diff1_kernel<<<256, 256, 0, stream>>>(Fr + (size_t)slot * NBD,
                                          Xr + (size_t)slot * NBD, parts);
    ctrl_kernel<<<1, 256, 0, stream>>>(parts, scal, flags);
    copy_kernel<<<CPB, 256, 0, stream>>>(Xr + (size_t)slot * NBD, best, N2, flags + 1);
  }

  // ---- final differentiable step + output projection ----
  f_eval(best, Fr, WihZT, bih_z, WhhZT, bhh_z, z0, GI, stream);
  out_kernel<<<BB, 256, 0, stream>>>(Fr, Wout, bout, out);
}